// MatrixMemory_67912022885191
// MI455X (gfx1250) — compile-verified
//
#include <hip/hip_runtime.h>

typedef __attribute__((ext_vector_type(2))) float v2f;
typedef __attribute__((ext_vector_type(4))) float v4f;
typedef __attribute__((ext_vector_type(8))) float v8f;

#define NB  2048   // batch
#define DK  256
#define DV  256

// One block per batch element; 512 threads = 16 wave32, one wave per 16-row tile.
__global__ __launch_bounds__(512)
void matrix_memory_fused(const float* __restrict__ state,   // (B, DV, DK)
                         const float* __restrict__ query,   // (B, DK)
                         const float* __restrict__ keyv,    // (B, DK)
                         const float* __restrict__ dout,    // (B, DV)
                         float* __restrict__ y,             // (B, DV)
                         float* __restrict__ dM)            // (B, DV, DK)
{
    const int b    = blockIdx.x;
    const int wave = threadIdx.x >> 5;   // 0..15
    const int lane = threadIdx.x & 31;   // wave32
    const int v0   = wave << 4;          // 16-row tile base

    const size_t bDK = (size_t)b * DK;
    const size_t bDV = (size_t)b * DV;

    // ------------------------------------------------------------------
    // y[v0..v0+15] = state[b, v0..v0+15, :] @ query[b, :]
    // via V_WMMA_F32_16X16X4_F32, B = q-chunk broadcast across 16 columns.
    // A 16x4 f32 layout (ISA 7.12.2): lane m (0..15) K={k,k+1}; lane m+16 K={k+2,k+3}
    // ------------------------------------------------------------------
    const int m    = lane & 15;
    const int kseg = (lane >> 4) << 1;   // 0 or 2

    const float* arow = state + (bDV + v0 + m) * (size_t)DK + kseg;
    const float* qrow = query + bDK + kseg;

    v8f acc = {0.f, 0.f, 0.f, 0.f, 0.f, 0.f, 0.f, 0.f};

    #pragma unroll 8
    for (int k = 0; k < DK; k += 4) {
        v2f a  = __builtin_nontemporal_load((const v2f*)(arow + k)); // 512MB stream: NT
        v2f qb = *(const v2f*)(qrow + k);                            // hot, tiny
        acc = __builtin_amdgcn_wmma_f32_16x16x4_f32(
                  /*neg_a=*/false, a, /*neg_b=*/false, qb,
                  /*c_mod=*/(short)0, acc, /*reuse_a=*/false, /*reuse_b=*/false);
    }

    // Every column of the 16x16 result is identical (B columns identical).
    // C/D layout: lane 0 -> N=0, M=0..7 in acc[0..7]; lane 16 -> N=0, M=8..15.
    if ((lane & 15) == 0) {
        float* yp = y + bDV + v0 + ((lane >> 4) << 3);
        #pragma unroll
        for (int i = 0; i < 8; ++i) yp[i] = acc[i];
    }

    // ------------------------------------------------------------------
    // dM[b, v0+r, :] = d_out[b, v0+r] * key[b, :]   (pure streaming write)
    // Each lane owns 8 key values (2 x float4); NT stores keep L2 clean.
    // ------------------------------------------------------------------
    v4f k0 = *(const v4f*)(keyv + bDK +       (lane << 2));
    v4f k1 = *(const v4f*)(keyv + bDK + 128 + (lane << 2));

    #pragma unroll 4
    for (int r = 0; r < 16; ++r) {
        const float g = dout[bDV + v0 + r];
        float* dst = dM + (bDV + v0 + r) * (size_t)DK;
        v4f o0 = k0 * g;
        v4f o1 = k1 * g;
        __builtin_nontemporal_store(o0, (v4f*)(dst +       (lane << 2)));
        __builtin_nontemporal_store(o1, (v4f*)(dst + 128 + (lane << 2)));
    }
}

extern "C" void kernel_launch(void* const* d_in, const int* in_sizes, int n_in,
                              void* d_out, int out_size, void* d_ws, size_t ws_size,
                              hipStream_t stream) {
    (void)in_sizes; (void)n_in; (void)d_ws; (void)ws_size; (void)out_size;

    const float* state = (const float*)d_in[0];  // (B, DV, DK)
    const float* query = (const float*)d_in[1];  // (B, DK)
    const float* keyv  = (const float*)d_in[2];  // (B, DK)
    const float* dout  = (const float*)d_in[3];  // (B, DV)

    float* y  = (float*)d_out;                   // first output: (B, DV)
    float* dM = y + (size_t)NB * DV;             // second output: (B, DV, DK)

    matrix_memory_fused<<<NB, 512, 0, stream>>>(state, query, keyv, dout, y, dM);
}